// HamiltonianLayer_49976239456647
// MI455X (gfx1250) — compile-verified
//
#include <hip/hip_runtime.h>

typedef float v2f __attribute__((ext_vector_type(2)));
typedef float v8f __attribute__((ext_vector_type(8)));

#define N_DIM    1024
#define NP       512
#define OUT_DIM  64
#define PPF      8
#define BATCH    128

// workspace layout (float offsets)
#define WS_PARTIAL 0            // 16*1024*2 = 32768 floats
#define WS_STATE   32768        // 1024*2
#define WS_VS      34816        // 512*2
#define WS_T       35840        // 128*512

// ---------------- K1: partial column sums of U (state = ones @ U / 32) ----
__global__ void __launch_bounds__(128) k_colsum_partial(
    const float* __restrict__ Ure, const float* __restrict__ Uim,
    float* __restrict__ ws) {
  const int colChunk = blockIdx.x & 7;     // 8 chunks of 128 cols
  const int rowChunk = blockIdx.x >> 3;    // 16 chunks of 64 rows
  const int n = colChunk * 128 + threadIdx.x;
  const int m0 = rowChunk * 64;
  float ar = 0.f, ai = 0.f;
  #pragma unroll 8
  for (int m = 0; m < 64; ++m) {
    ar += Ure[(m0 + m) * N_DIM + n];       // coalesced across threads
    ai += Uim[(m0 + m) * N_DIM + n];
  }
  float* partial = ws + WS_PARTIAL;
  partial[(rowChunk * N_DIM + n) * 2 + 0] = ar;
  partial[(rowChunk * N_DIM + n) * 2 + 1] = ai;
}

// ---------------- K2: reduce partials -> state[n] (complex) --------------
__global__ void __launch_bounds__(256) k_state_reduce(float* __restrict__ ws) {
  const int n = blockIdx.x * blockDim.x + threadIdx.x;   // 1024 total
  const float* partial = ws + WS_PARTIAL;
  float ar = 0.f, ai = 0.f;
  #pragma unroll
  for (int r = 0; r < 16; ++r) {
    ar += partial[(r * N_DIM + n) * 2 + 0];
    ai += partial[(r * N_DIM + n) * 2 + 1];
  }
  float* st = ws + WS_STATE;
  st[n * 2 + 0] = ar * (1.0f / 32.0f);     // sent = 1/sqrt(1024)
  st[n * 2 + 1] = ai * (1.0f / 32.0f);
}

// ---------------- K3: vs[p] = u[p] * sum_n conj(st[n]) st[perm] w[p,n] ----
__global__ void __launch_bounds__(128) k_state_pauli(
    const int* __restrict__ perm, const float* __restrict__ sgn,
    float* __restrict__ ws) {
  const int p = blockIdx.x;                // 512 blocks
  const int t = threadIdx.x;               // 128 threads
  const float* st = ws + WS_STATE;
  float qre = 0.f, qim = 0.f;
  #pragma unroll
  for (int k = 0; k < 8; ++k) {
    const int n = t + k * 128;
    const int idx = p * N_DIM + n;
    const int pm = perm[idx];                          // coalesced
    const float sr = sgn[idx * 2], si = sgn[idx * 2 + 1];
    const float w = sr + si;                           // one of them is 0
    const float str = st[n * 2], sti = st[n * 2 + 1];
    const float gr = st[pm * 2], gi = st[pm * 2 + 1];  // L2-resident gather
    qre += w * (str * gr + sti * gi);                  // Re(conj(st)*g)
    qim += w * (str * gi - sti * gr);                  // Im(conj(st)*g)
  }
  __shared__ float rre[128], rim[128];
  rre[t] = qre; rim[t] = qim;
  __syncthreads();
  for (int s = 64; s > 0; s >>= 1) {
    if (t < s) { rre[t] += rre[t + s]; rim[t] += rim[t + s]; }
    __syncthreads();
  }
  if (t == 0) {
    const bool isImag = (sgn[p * N_DIM * 2] == 0.0f);  // phase = signs[p,0]
    float* vs = ws + WS_VS;
    vs[p * 2 + 0] = isImag ? -rim[0] : rre[0];         // u = i or 1
    vs[p * 2 + 1] = isImag ?  rre[0] : rim[0];
  }
}

// ---------------- K4: T[b,p] via WMMA f32 16x16x4 reductions -------------
// Block = (batch-group of 8, P-slab of 64 Paulis). 8 waves: wave handles a
// 16-Pauli subgroup for 4 batch rows (4 independent WMMA accumulator chains).
// A[p,k] = x[b, perm[p,n0+k]] * w[p,n0+k]  (16 Paulis x K=4)
// B[k,j] = x[b, n0+k] broadcast to all 16 columns -> C columns identical.
__global__ void __launch_bounds__(256) k_xpauli_wmma(
    const float* __restrict__ x, const int* __restrict__ perm,
    const float* __restrict__ sgn, float* __restrict__ ws) {
  __shared__ float xs[8 * N_DIM];          // 32 KB: 8 batch rows
  const int bg = blockIdx.x >> 3;          // 0..15 batch groups
  const int pg = blockIdx.x & 7;           // 0..7 == P slab
  const int tid = threadIdx.x;
  for (int i = tid; i < 8 * N_DIM; i += 256)
    xs[i] = x[(bg * 8 + (i >> 10)) * N_DIM + (i & 1023)];
  __syncthreads();

  const int wid  = tid >> 5;
  const int lane = tid & 31;
  const int sub  = wid >> 1;               // 16-Pauli subgroup 0..3
  const int bh   = wid & 1;                // batch half (4 rows each)
  const int pbase = pg * 64 + sub * 16;
  const int pA   = pbase + (lane & 15);    // A row (M) for this lane
  const int kOff = (lane < 16) ? 0 : 2;    // ISA: A lanes 0-15 hold K=0,1; 16-31 K=2,3
  const int kB0  = (lane < 16) ? 0 : 2;    // B rows per C/D striping analogy
  const float* xr0 = &xs[(bh * 4 + 0) * N_DIM];
  const float* xr1 = &xs[(bh * 4 + 1) * N_DIM];
  const float* xr2 = &xs[(bh * 4 + 2) * N_DIM];
  const float* xr3 = &xs[(bh * 4 + 3) * N_DIM];

  v8f c0 = {}, c1 = {}, c2 = {}, c3 = {};
  for (int n0 = 0; n0 < N_DIM; n0 += 4) {
    const int base = pA * N_DIM + n0 + kOff;                 // even -> aligned
    const int2   pm = *(const int2*)(perm + base);           // b64 load
    const float4 sg = *(const float4*)(sgn + 2 * base);      // b128 load
    const float w0 = sg.x + sg.y;
    const float w1 = sg.z + sg.w;
    const int nB0 = n0 + kB0, nB1 = n0 + kB0 + 1;
    v2f A, B;
    A.x = xr0[pm.x] * w0; A.y = xr0[pm.y] * w1;
    B.x = xr0[nB0];       B.y = xr0[nB1];
    c0 = __builtin_amdgcn_wmma_f32_16x16x4_f32(false, A, false, B, (short)0, c0, false, false);
    A.x = xr1[pm.x] * w0; A.y = xr1[pm.y] * w1;
    B.x = xr1[nB0];       B.y = xr1[nB1];
    c1 = __builtin_amdgcn_wmma_f32_16x16x4_f32(false, A, false, B, (short)0, c1, false, false);
    A.x = xr2[pm.x] * w0; A.y = xr2[pm.y] * w1;
    B.x = xr2[nB0];       B.y = xr2[nB1];
    c2 = __builtin_amdgcn_wmma_f32_16x16x4_f32(false, A, false, B, (short)0, c2, false, false);
    A.x = xr3[pm.x] * w0; A.y = xr3[pm.y] * w1;
    B.x = xr3[nB0];       B.y = xr3[nB1];
    c3 = __builtin_amdgcn_wmma_f32_16x16x4_f32(false, A, false, B, (short)0, c3, false, false);
  }
  // C/D layout: VGPR r = rows M=r (lanes 0-15) and M=r+8 (lanes 16-31);
  // all 16 columns equal, so lane 0 / lane 16 hold the results.
  if (lane == 0 || lane == 16) {
    float* T = ws + WS_T;
    const int pr = pbase + ((lane == 16) ? 8 : 0);
    const int brow = bg * 8 + bh * 4;
    #pragma unroll
    for (int r = 0; r < 8; ++r) {
      T[(brow + 0) * NP + pr + r] = c0[r];
      T[(brow + 1) * NP + pr + r] = c1[r];
      T[(brow + 2) * NP + pr + r] = c2[r];
      T[(brow + 3) * NP + pr + r] = c3[r];
    }
  }
}

// ---------------- K5: combine P-terms, Re(x_p * s_p), BN + ReLU ----------
__global__ void __launch_bounds__(128) k_finalize(
    const float* __restrict__ sgn, const float* __restrict__ rw,
    const float* __restrict__ gamma, const float* __restrict__ beta,
    const float* __restrict__ ws, float* __restrict__ out) {
  const int f = blockIdx.x;                // 64 blocks, one per feature
  const int b = threadIdx.x;               // 128 threads, one per batch row
  const float* T  = ws + WS_T;
  const float* vs = ws + WS_VS;
  float accRe = 0.f, accIm = 0.f, sRe = 0.f, sIm = 0.f;
  #pragma unroll
  for (int P = 0; P < PPF; ++P) {
    const int p = P * OUT_DIM + f;
    const float t = rw[P * OUT_DIM + f] * T[b * NP + p];
    const bool isImag = (sgn[p * N_DIM * 2] == 0.0f);
    accRe += isImag ? 0.f : t;             // x_p contribution (u = 1 or i)
    accIm += isImag ? t : 0.f;
    sRe += vs[p * 2];                      // s_p[f] = sum_P vs[P*64+f]
    sIm += vs[p * 2 + 1];
  }
  const float o = accRe * sRe - accIm * sIm;   // Re(x_p * s_p)
  __shared__ float s1[128], s2[128];
  s1[b] = o; s2[b] = o * o;
  __syncthreads();
  for (int s = 64; s > 0; s >>= 1) {
    if (b < s) { s1[b] += s1[b + s]; s2[b] += s2[b + s]; }
    __syncthreads();
  }
  const float m  = s1[0] * (1.0f / BATCH);
  const float va = s2[0] * (1.0f / BATCH) - m * m;    // biased var (ddof=0)
  const float y = gamma[f] * (o - m) * rsqrtf(va + 1e-5f) + beta[f];
  out[b * OUT_DIM + f] = fmaxf(y, 0.0f);
}

extern "C" void kernel_launch(void* const* d_in, const int* in_sizes, int n_in,
                              void* d_out, int out_size, void* d_ws, size_t ws_size,
                              hipStream_t stream) {
  const float* x     = (const float*)d_in[0];
  const int*   perm  = (const int*)  d_in[1];
  const float* sgn   = (const float*)d_in[2];   // complex64 interleaved re,im
  const float* rw    = (const float*)d_in[3];
  const float* Ure   = (const float*)d_in[4];
  const float* Uim   = (const float*)d_in[5];
  const float* gamma = (const float*)d_in[6];
  const float* beta  = (const float*)d_in[7];
  float* ws  = (float*)d_ws;
  float* out = (float*)d_out;

  k_colsum_partial<<<128, 128, 0, stream>>>(Ure, Uim, ws);
  k_state_reduce <<<4,   256, 0, stream>>>(ws);
  k_state_pauli  <<<NP,  128, 0, stream>>>(perm, sgn, ws);
  k_xpauli_wmma  <<<128, 256, 0, stream>>>(x, perm, sgn, ws);
  k_finalize     <<<OUT_DIM, 128, 0, stream>>>(sgn, rw, gamma, beta, ws, out);
}